// POSEncoder1_20968030339747
// MI455X (gfx1250) — compile-verified
//
#include <hip/hip_runtime.h>
#include <hip/hip_bf16.h>

#define NUM_EXPERTS 45
#define EMBED_DIM   256
#define NUM_TOKENS  4096
#define TILE_M      16
#define XS_STRIDE   (EMBED_DIM + 4)   // pad LDS rows to avoid 16-way bank conflicts

typedef __attribute__((ext_vector_type(2))) float v2f;
typedef __attribute__((ext_vector_type(8))) float v8f;

// ---------------------------------------------------------------------------
// Kernel 1: bucket tokens by expert tag (histogram -> prefix -> scatter),
// and zero d_out (ReLU output >= 0, so 0.0f is the identity for the max).
// Bucket order within an expert is atomic-order dependent, but every token's
// y-row is computed independently and max() is exactly commutative, so the
// final output is deterministic.
// ---------------------------------------------------------------------------
__global__ __launch_bounds__(1024) void pos_bucket_kernel(
    const int* __restrict__ tags,
    int* __restrict__ counts,
    int* __restrict__ offsets,
    int* __restrict__ order,
    float* __restrict__ out)
{
    __shared__ int sc[NUM_EXPERTS];
    __shared__ int so[NUM_EXPERTS];
    const int tid = threadIdx.x;

    if (tid < NUM_EXPERTS) sc[tid] = 0;
    if (tid < EMBED_DIM)   out[tid] = 0.0f;
    __syncthreads();

    for (int i = tid; i < NUM_TOKENS; i += 1024)
        atomicAdd(&sc[tags[i]], 1);
    __syncthreads();

    if (tid == 0) {
        int run = 0;
        for (int e = 0; e < NUM_EXPERTS; ++e) { so[e] = run; run += sc[e]; }
    }
    __syncthreads();

    if (tid < NUM_EXPERTS) {
        counts[tid]  = sc[tid];
        offsets[tid] = so[tid];
        sc[tid]      = so[tid];   // reuse as running scatter cursor
    }
    __syncthreads();

    for (int i = tid; i < NUM_TOKENS; i += 1024) {
        int p = atomicAdd(&sc[tags[i]], 1);
        order[p] = i;
    }
}

// ---------------------------------------------------------------------------
// Kernel 2: per (expert, 16-token tile) block. 256 threads = 8 waves.
// Each wave owns two 16-wide N-tiles; K=256 swept with v_wmma_f32_16x16x4_f32.
//   A (16x4 f32, 2 VGPRs): lane l<16 -> M=l, K={k0,k0+1}; lanes 16..31 -> K={k0+2,k0+3}
//   B (4x16 f32, 2 VGPRs): lane l<16 -> N=l, K={k0,k0+1}; lanes 16..31 -> K={k0+2,k0+3}
//   C/D (16x16 f32, 8 VGPRs): VGPR r -> M = r + 8*(lane>=16), N = lane&15
// ---------------------------------------------------------------------------
__global__ __launch_bounds__(256) void pos_gemm_wmma_kernel(
    const float* __restrict__ x,
    const float* __restrict__ W,
    const float* __restrict__ bias,
    const int*  __restrict__ counts,
    const int*  __restrict__ offsets,
    const int*  __restrict__ order,
    float* __restrict__ out)
{
    const int e = blockIdx.x;
    const int t = blockIdx.y;
    const int cnt = counts[e];
    if (t * TILE_M >= cnt) return;               // block-uniform exit

    __shared__ int   sidx[TILE_M];
    __shared__ float xs[TILE_M][XS_STRIDE];

    const int tid  = threadIdx.x;
    const int base = offsets[e] + t * TILE_M;
    const int rem  = cnt - t * TILE_M;

    if (tid < TILE_M)
        sidx[tid] = (tid < rem) ? order[base + tid] : -1;
    __syncthreads();

    // Gather 16 token rows of x into LDS (coalesced 1KB row loads).
    for (int m = 0; m < TILE_M; ++m) {
        const int row = sidx[m];
        xs[m][tid] = (row >= 0) ? x[(size_t)row * EMBED_DIM + tid] : 0.0f;
    }
    __syncthreads();

    const int wave  = tid >> 5;          // 0..7
    const int lane  = tid & 31;
    const int mloc  = lane & 15;         // M (for A) / N offset (for B and C)
    const int khalf = (lane >> 4) * 2;   // K sub-offset per lane half

    const float* We = W + (size_t)e * EMBED_DIM * EMBED_DIM;
    unsigned int* outu = (unsigned int*)out;

    #pragma unroll
    for (int nt = 0; nt < 2; ++nt) {
        const int n0 = (wave * 2 + nt) * 16;
        const int n  = n0 + mloc;
        const float* wrow = We + (size_t)n * EMBED_DIM;   // row n of W[e]

        v8f acc = {};
        for (int k0 = 0; k0 < EMBED_DIM; k0 += 4) {
            if ((k0 & 63) == 0)
                __builtin_prefetch(wrow + k0 + 64, 0, 0);  // global_prefetch_b8

            const int kb = k0 + khalf;
            // A fragment from LDS (8-byte aligned -> ds_load_b64)
            const float2 av = *(const float2*)&xs[mloc][kb];
            // B fragment = W[e][n][kb..kb+1] (8-byte global load, L2-hot)
            const float2 bv = *(const float2*)(wrow + kb);

            v2f a = { av.x, av.y };
            v2f b = { bv.x, bv.y };
            acc = __builtin_amdgcn_wmma_f32_16x16x4_f32(
                /*neg_a=*/false, a, /*neg_b=*/false, b,
                /*c_mod=*/(short)0, acc, /*reuse_a=*/false, /*reuse_b=*/false);
        }

        // Epilogue: bias + ReLU + elementwise max over valid tokens.
        // All values >= 0 after ReLU, so uint atomicMax on the float bit
        // pattern is an exact, order-independent float max.
        const float bn = bias[e * EMBED_DIM + n];
        #pragma unroll
        for (int r = 0; r < 8; ++r) {
            const int m = r + (lane >> 4) * 8;
            if (sidx[m] >= 0) {
                float v = acc[r] + bn;
                v = v > 0.0f ? v : 0.0f;
                atomicMax(&outu[n], __float_as_uint(v));
            }
        }
    }
}

// ---------------------------------------------------------------------------
extern "C" void kernel_launch(void* const* d_in, const int* in_sizes, int n_in,
                              void* d_out, int out_size, void* d_ws, size_t ws_size,
                              hipStream_t stream) {
    const float* x    = (const float*)d_in[0];   // [4096, 256] f32
    const int*   tags = (const int*)  d_in[1];   // [4096] i32
    const float* W    = (const float*)d_in[2];   // [45, 256, 256] f32
    const float* bias = (const float*)d_in[3];   // [45, 256] f32
    float* out = (float*)d_out;                  // [256] f32

    int* wsI     = (int*)d_ws;
    int* counts  = wsI;          // 45 (padded to 64)
    int* offsets = wsI + 64;     // 45 (padded to 64)
    int* order   = wsI + 128;    // 4096

    pos_bucket_kernel<<<1, 1024, 0, stream>>>(tags, counts, offsets, order, out);

    dim3 grid(NUM_EXPERTS, NUM_TOKENS / TILE_M);   // worst case: all tokens one expert
    pos_gemm_wmma_kernel<<<grid, 256, 0, stream>>>(x, W, bias,
                                                   counts, offsets, order, out);
}